// DataReuploadingEncoder_91199335563808
// MI455X (gfx1250) — compile-verified
//
#include <hip/hip_runtime.h>
#include <math.h>

// Quantum data-reuploading encoder, 4 qubits, 3 layers, B = 1M samples.
// Shared per-layer unitary (Rot gates + CNOT ring) precomputed as a 16x16
// complex matrix and applied to 16-sample tiles via V_WMMA_F32_16X16X4_F32.

#define NQ 4
#define NL 3

typedef __attribute__((ext_vector_type(2))) float v2f;
typedef __attribute__((ext_vector_type(8))) float v8f;

// ---------------------------------------------------------------------------
// d_ws layout (6 KB): U tiles stored pre-swizzled in WMMA A-operand layout.
//   slot(l, reim, t, j, lane) = ((l*2 + reim)*8 + (t*2 + j))*32 + lane
//   holds U_layer[m][k] with m = lane&15, k = 4*t + j + 2*(lane>=16)
// ---------------------------------------------------------------------------

__global__ void build_layer_unitaries_kernel(const float* __restrict__ w,
                                             float* __restrict__ ws) {
  // U_layer = P_cnot_ring * (Rot0 (x) Rot1 (x) Rot2 (x) Rot3)
  // amp-index bit of qubit q is bit (3-q)  (matches [B,2,2,2,2] C-order).
  for (int idx = threadIdx.x; idx < NL * 256; idx += blockDim.x) {
    int l = idx >> 8;
    int e = idx & 255;
    int i = e >> 4;   // output row (amp index after CNOT ring)
    int b = e & 15;   // input col (amp index)
    // Row i of U_layer = row j of the Rot tensor product, where j is obtained
    // by walking the CNOT ring backwards: (Cs)_i = s_{g(i)},
    // g_cnot(c,t)(i) = i ^ (bit(3-c)(i) ? 1<<(3-t) : 0)
    int j = i;
    j = (j & 1) ? (j ^ 8) : j;  // CNOT(3,0)
    j = (j & 2) ? (j ^ 1) : j;  // CNOT(2,3)
    j = (j & 4) ? (j ^ 2) : j;  // CNOT(1,2)
    j = (j & 8) ? (j ^ 4) : j;  // CNOT(0,1)
    float pr = 1.0f, pim = 0.0f;
#pragma unroll
    for (int q = 0; q < 4; ++q) {
      const float* wq = w + (l * 4 + q) * 3;
      float phi = wq[0], th = wq[1], om = wq[2];
      // Rot(phi,theta,omega) = RZ(om) RY(th) RZ(phi):
      //   m00 = c e^{-i a}, m01 = -s e^{+i bb}, m10 = s e^{-i bb}, m11 = c e^{+i a}
      float c = cosf(0.5f * th), s = sinf(0.5f * th);
      float a = 0.5f * (phi + om), bb = 0.5f * (phi - om);
      int rb = (j >> (3 - q)) & 1;
      int cb = (b >> (3 - q)) & 1;
      float er, ei;
      if (rb == 0 && cb == 0)      { er =  c * cosf(a);  ei = -c * sinf(a);  }
      else if (rb == 0 && cb == 1) { er = -s * cosf(bb); ei = -s * sinf(bb); }
      else if (rb == 1 && cb == 0) { er =  s * cosf(bb); ei = -s * sinf(bb); }
      else                         { er =  c * cosf(a);  ei =  c * sinf(a);  }
      float nr = pr * er - pim * ei;
      float ni = pr * ei + pim * er;
      pr = nr; pim = ni;
    }
    // scatter into WMMA A-operand register layout
    int t = b >> 2, rem = b & 3;
    int lane = i + ((rem >= 2) ? 16 : 0);
    int jj = rem & 1;
    int slot = (t * 2 + jj) * 32 + lane;
    ws[(l * 2 + 0) * 256 + slot] = pr;
    ws[(l * 2 + 1) * 256 + slot] = pim;
  }
}

__device__ __forceinline__ v8f wmma_f32(v2f a, v2f b, v8f c) {
  // D = A(16x4 f32) x B(4x16 f32) + C(16x16 f32)
  return __builtin_amdgcn_wmma_f32_16x16x4_f32(
      false, a, false, b, (short)0, c, false, false);
}

// ---------------------------------------------------------------------------
// Main kernel: one wave32 simulates 16 samples.
// State layout = WMMA C/D layout: lane L holds sample (L&15), amplitudes
// r + 8*(L>=16) for r = VGPR 0..7 (re[] and im[] each v8f).
// ---------------------------------------------------------------------------
__global__ void __launch_bounds__(256)
quantum_reupload_kernel(const float* __restrict__ x,
                        const float* __restrict__ scaling,
                        const float* __restrict__ U,
                        float* __restrict__ out, int B) {
  const int lane = threadIdx.x & 31;
  const bool lo = lane < 16;
  const int wave = blockIdx.x * (blockDim.x >> 5) + (threadIdx.x >> 5);
  const int s = wave * 16 + (lane & 15);
  const int sc = (s < B) ? s : (B - 1);

  float4 xv = ((const float4*)x)[sc];
  float xt[4] = { tanhf(xv.x), tanhf(xv.y), tanhf(xv.z), tanhf(xv.w) };

  v8f re = {};
  v8f im = {};
  re[0] = lo ? 1.0f : 0.0f;  // |0000> : amplitude 0 lives in VGPR0, lanes 0-15

  const float kPi = 3.14159265358979323846f;

#pragma unroll
  for (int l = 0; l < NL; ++l) {
    // ---- per-sample data-encoding gates: m = RZ(ang/2) @ RY(ang) ----------
#pragma unroll
    for (int q = 0; q < 4; ++q) {
      float ang = xt[q] * scaling[l * 4 + q] * kPi;
      float c, sn, cp, sp;
      __sincosf(0.5f * ang, &sn, &c);    // RY(ang) entries
      __sincosf(0.25f * ang, &sp, &cp);  // RZ(ang/2) phase e^{-i ang/4}
      // m00 = ( cp*c, -sp*c)   m01 = (-cp*sn,  sp*sn)
      // m10 = ( cp*sn, sp*sn)  m11 = ( cp*c,   sp*c)
      if (q == 0) {
        // qubit 0 = amp bit 3: partner lives in the other lane half
        float cSelfR = cp * c;
        float cSelfI = lo ? -sp * c : sp * c;
        float cOthR  = lo ? -cp * sn : cp * sn;
        float cOthI  = sp * sn;
#pragma unroll
        for (int r = 0; r < 8; ++r) {
          float orr = __shfl_xor(re[r], 16, 32);
          float oii = __shfl_xor(im[r], 16, 32);
          float nr = cSelfR * re[r] - cSelfI * im[r] + cOthR * orr - cOthI * oii;
          float ni = cSelfR * im[r] + cSelfI * re[r] + cOthR * oii + cOthI * orr;
          re[r] = nr; im[r] = ni;
        }
      } else {
        const int mask = 8 >> q;  // VGPR-local amplitude bit
        float m00r =  cp * c,  m00i = -sp * c;
        float m01r = -cp * sn, m01i =  sp * sn;
        float m10r =  cp * sn, m10i =  sp * sn;
        float m11r =  cp * c,  m11i =  sp * c;
#pragma unroll
        for (int r = 0; r < 8; ++r) {
          if (r & mask) continue;
          const int r2 = r | mask;
          float ar = re[r], ai = im[r], br = re[r2], bi = im[r2];
          re[r]  = m00r * ar - m00i * ai + m01r * br - m01i * bi;
          im[r]  = m00r * ai + m00i * ar + m01r * bi + m01i * br;
          re[r2] = m10r * ar - m10i * ai + m11r * br - m11i * bi;
          im[r2] = m10r * ai + m10i * ar + m11r * bi + m11i * br;
        }
      }
    }

    // ---- shared layer unitary via WMMA: new = (Ur + i Ui)(Re + i Im) ------
    // A tiles (pre-swizzled in ws)
    v2f aRe[4], aIm[4], aImN[4];
    const float* ur = U + (l * 2 + 0) * 256;
    const float* ui = U + (l * 2 + 1) * 256;
#pragma unroll
    for (int t = 0; t < 4; ++t) {
      aRe[t]  = v2f{ ur[(t * 2 + 0) * 32 + lane], ur[(t * 2 + 1) * 32 + lane] };
      aIm[t]  = v2f{ ui[(t * 2 + 0) * 32 + lane], ui[(t * 2 + 1) * 32 + lane] };
      aImN[t] = -aIm[t];  // f32 WMMA has no A-negate modifier; materialize -Ui
    }

    // B tiles: convert C/D state layout -> 4x16 B-operand tiles
    float sr[8], si[8];
#pragma unroll
    for (int r = 0; r < 8; ++r) {
      sr[r] = __shfl_xor(re[r], 16, 32);
      si[r] = __shfl_xor(im[r], 16, 32);
    }
    v2f bRe[4], bIm[4];
    bRe[0] = v2f{ lo ? re[0] : sr[2], lo ? re[1] : sr[3] };
    bRe[1] = v2f{ lo ? re[4] : sr[6], lo ? re[5] : sr[7] };
    bRe[2] = v2f{ lo ? sr[0] : re[2], lo ? sr[1] : re[3] };
    bRe[3] = v2f{ lo ? sr[4] : re[6], lo ? sr[5] : re[7] };
    bIm[0] = v2f{ lo ? im[0] : si[2], lo ? im[1] : si[3] };
    bIm[1] = v2f{ lo ? im[4] : si[6], lo ? im[5] : si[7] };
    bIm[2] = v2f{ lo ? si[0] : im[2], lo ? si[1] : im[3] };
    bIm[3] = v2f{ lo ? si[4] : im[6], lo ? si[5] : im[7] };

    v8f dRe = {};
    v8f dIm = {};
#pragma unroll
    for (int t = 0; t < 4; ++t) dRe = wmma_f32(aRe[t],  bRe[t], dRe);
#pragma unroll
    for (int t = 0; t < 4; ++t) dRe = wmma_f32(aImN[t], bIm[t], dRe);
#pragma unroll
    for (int t = 0; t < 4; ++t) dIm = wmma_f32(aIm[t],  bRe[t], dIm);
#pragma unroll
    for (int t = 0; t < 4; ++t) dIm = wmma_f32(aRe[t],  bIm[t], dIm);
    re = dRe;
    im = dIm;
  }

  // ---- Pauli-Z expectation values --------------------------------------
  float ev[4];
#pragma unroll
  for (int w = 0; w < 4; ++w) {
    const int bit = 3 - w;
    float p = 0.0f;
#pragma unroll
    for (int r = 0; r < 8; ++r) {
      int amp = r + (lo ? 0 : 8);
      float pr = re[r] * re[r] + im[r] * im[r];
      p += ((amp >> bit) & 1) ? -pr : pr;
    }
    p += __shfl_xor(p, 16, 32);  // combine the two amplitude halves
    ev[w] = p;
  }
  if (lo && s < B) {
    ((float4*)out)[s] = make_float4(ev[0], ev[1], ev[2], ev[3]);
  }
}

extern "C" void kernel_launch(void* const* d_in, const int* in_sizes, int n_in,
                              void* d_out, int out_size, void* d_ws, size_t ws_size,
                              hipStream_t stream) {
  const float* x       = (const float*)d_in[0];  // [B,4] f32
  const float* weights = (const float*)d_in[1];  // [3,4,3] f32
  const float* scaling = (const float*)d_in[2];  // [3,4] f32
  float* out = (float*)d_out;                    // [B,4] f32
  float* U   = (float*)d_ws;                     // 1536 f32 (6 KB) of A-tiles

  const int B = in_sizes[0] / 4;

  build_layer_unitaries_kernel<<<1, 256, 0, stream>>>(weights, U);

  const int tiles = (B + 15) / 16;
  const int wavesPerBlock = 8;  // 256 threads = 8 wave32
  const int blocks = (tiles + wavesPerBlock - 1) / wavesPerBlock;
  quantum_reupload_kernel<<<blocks, 256, 0, stream>>>(x, scaling, U, out, B);
}